// VisionMamba3Block_79001628443007
// MI455X (gfx1250) — compile-verified
//
#include <hip/hip_runtime.h>
#include <math.h>

#define BATCH   4
#define DIMC    384
#define HH      24
#define WWID    24
#define HWX     576
#define LSEQ    576
#define TOK     (BATCH*LSEQ)     // 2304
#define DINNER  768
#define DSTATE  64
#define NHEADS  12
#define HEADDIM 64
#define NPROJ   1676             // 2*DINNER + 2*DSTATE + NHEADS
#define XBCW    896              // DINNER + 2*DSTATE
#define MLPHID  768
#define GATEHID 96
#define GROUPS  8
#define CPG     (DIMC/GROUPS)    // 48

typedef __attribute__((ext_vector_type(2))) float v2f;
typedef __attribute__((ext_vector_type(8))) float v8f;

__device__ __forceinline__ float gelu_f(float x){ return 0.5f*x*(1.0f+erff(x*0.70710678118654752f)); }
__device__ __forceinline__ float silu_f(float x){ return x/(1.0f+expf(-x)); }

// ---------------------------------------------------------------------------
// Generic fp32 WMMA GEMM, 16(M) x 64(N) tile per wave, 4 waves per block.
//   BMODE==0 : C[M,N] = A[M,K] * X[K,N]      (X row-major K x N, strided b32 loads)
//   BMODE==1 : C[M,N] = A[M,K] * W[N,K]^T    (W row-major N x K, contiguous b64 loads)
// Each K-step loads one A fragment and feeds FOUR independent
// V_WMMA_F32_16X16X4_F32 accumulation chains -> 4x A reuse + WMMA ILP.
// A/B fragment pairs are contiguous & 8B-aligned (ka even, K % 4 == 0), so
// BMODE==1 uses single global_load_b64 per fragment. fp32 WMMA keeps exact
// reference precision; the whole problem is L2-resident so matrix peak is not
// the limiter.
// ---------------------------------------------------------------------------
template<int BMODE>
__global__ void k_gemm(const float* __restrict__ A, const float* __restrict__ Bm,
                       const float* __restrict__ bias, float* __restrict__ C,
                       int M, int N, int K, long sA, long sB, long sC)
{
  const float* Ab = A + (long)blockIdx.z * sA;
  const float* Bb = Bm + (long)blockIdx.z * sB;
  float*       Cb = C + (long)blockIdx.z * sC;
  int tilesN = (N + 63) >> 6;
  long nTiles = (long)((M + 15) >> 4) * tilesN;
  long tid = (long)blockIdx.x * (blockDim.x >> 5) + (threadIdx.x >> 5);
  if (tid >= nTiles) return;                 // wave-uniform: EXEC stays all-ones
  int tm = (int)(tid / tilesN), tn = (int)(tid % tilesN);
  int lane = threadIdx.x & 31;
  int half = lane >> 4, r = lane & 15;
  int m = tm*16 + r;
  bool mval = (m < M);
  int mc = mval ? m : 0;
  const float* arow = Ab + (long)mc * K;

  int nj[4]; bool nvalj[4]; int ncj[4];
  #pragma unroll
  for (int j = 0; j < 4; ++j) {
    nj[j] = tn*64 + j*16 + r;
    nvalj[j] = (nj[j] < N);
    ncj[j] = nvalj[j] ? nj[j] : 0;
  }
  const float* wrow0 = Bb + (long)ncj[0] * K;   // BMODE==1 row bases
  const float* wrow1 = Bb + (long)ncj[1] * K;
  const float* wrow2 = Bb + (long)ncj[2] * K;
  const float* wrow3 = Bb + (long)ncj[3] * K;

  v8f acc0 = {0.f,0.f,0.f,0.f,0.f,0.f,0.f,0.f};
  v8f acc1 = acc0, acc2 = acc0, acc3 = acc0;

  for (int k0 = 0; k0 < K; k0 += 4) {
    int ka = k0 + 2*half;                    // frag layout: reg0->K=ka, reg1->K=ka+1
    v2f af = *(const v2f*)(arow + ka);       // global_load_b64
    if (!mval) { af.x = 0.f; af.y = 0.f; }
    v2f bf0, bf1, bf2, bf3;
    if (BMODE == 0) {
      const float* brow0 = Bb + (long)ka*N;
      const float* brow1 = brow0 + N;
      bf0.x = brow0[ncj[0]]; bf0.y = brow1[ncj[0]];
      bf1.x = brow0[ncj[1]]; bf1.y = brow1[ncj[1]];
      bf2.x = brow0[ncj[2]]; bf2.y = brow1[ncj[2]];
      bf3.x = brow0[ncj[3]]; bf3.y = brow1[ncj[3]];
    } else {
      bf0 = *(const v2f*)(wrow0 + ka);       // global_load_b64 each
      bf1 = *(const v2f*)(wrow1 + ka);
      bf2 = *(const v2f*)(wrow2 + ka);
      bf3 = *(const v2f*)(wrow3 + ka);
    }
    if (!nvalj[0]) { bf0.x = 0.f; bf0.y = 0.f; }
    if (!nvalj[1]) { bf1.x = 0.f; bf1.y = 0.f; }
    if (!nvalj[2]) { bf2.x = 0.f; bf2.y = 0.f; }
    if (!nvalj[3]) { bf3.x = 0.f; bf3.y = 0.f; }
    acc0 = __builtin_amdgcn_wmma_f32_16x16x4_f32(false, af, false, bf0, (short)0, acc0, false, false);
    acc1 = __builtin_amdgcn_wmma_f32_16x16x4_f32(false, af, false, bf1, (short)0, acc1, false, false);
    acc2 = __builtin_amdgcn_wmma_f32_16x16x4_f32(false, af, false, bf2, (short)0, acc2, false, false);
    acc3 = __builtin_amdgcn_wmma_f32_16x16x4_f32(false, af, false, bf3, (short)0, acc3, false, false);
  }

  v8f* accs[4] = {&acc0, &acc1, &acc2, &acc3};
  #pragma unroll
  for (int j = 0; j < 4; ++j) {
    if (!nvalj[j]) continue;
    float bv = (bias != nullptr) ? bias[ncj[j]] : 0.f;
    v8f a = *accs[j];
    #pragma unroll
    for (int v = 0; v < 8; ++v) {
      int mr = tm*16 + v + 8*half;           // D layout: VGPR v -> M = v + 8*half
      if (mr < M) Cb[(long)mr*N + nj[j]] = a[v] + bv;
    }
  }
}

// --------------------------- local enhancer --------------------------------
// GroupNorm (+affine) + exact GELU, in place on (B, C, HW) buffer. 1 block per (b,g).
__global__ void k_gn_gelu(float* buf, const float* __restrict__ w, const float* __restrict__ b)
{
  int bg = blockIdx.x, bi = bg / GROUPS, g = bg % GROUPS;
  float* base = buf + (size_t)bi*DIMC*HWX + (size_t)g*CPG*HWX;
  const int NEL = CPG*HWX;
  __shared__ float s1[256], s2[256];
  float a = 0.f, a2 = 0.f;
  for (int i = threadIdx.x; i < NEL; i += 256) { float v = base[i]; a += v; a2 += v*v; }
  s1[threadIdx.x] = a; s2[threadIdx.x] = a2; __syncthreads();
  for (int st = 128; st > 0; st >>= 1) {
    if ((int)threadIdx.x < st) { s1[threadIdx.x] += s1[threadIdx.x+st]; s2[threadIdx.x] += s2[threadIdx.x+st]; }
    __syncthreads();
  }
  float m = s1[0] / NEL;
  float inv = rsqrtf(s2[0]/NEL - m*m + 1e-5f);
  for (int i = threadIdx.x; i < NEL; i += 256) {
    int c = g*CPG + i/HWX;
    base[i] = gelu_f((base[i]-m)*inv*w[c] + b[c]);
  }
}

// Depthwise 2D conv, same padding, k in {1,3,5}. 1 block per (b,c), 576 threads.
__global__ void k_dwconv(const float* __restrict__ in, const float* __restrict__ wt,
                         float* __restrict__ out, int k)
{
  int bc = blockIdx.x, c = bc % DIMC;
  int t = threadIdx.x, y = t / WWID, x = t % WWID;
  const float* src = in + (size_t)bc*HWX;
  const float* wk  = wt + (size_t)c*k*k;
  int r = k/2; float acc = 0.f;
  for (int kh = 0; kh < k; ++kh) { int iy = y+kh-r; if (iy < 0 || iy >= HH) continue;
    for (int kw = 0; kw < k; ++kw) { int ix = x+kw-r; if (ix < 0 || ix >= WWID) continue;
      acc += wk[kh*k+kw]*src[iy*WWID+ix]; } }
  out[(size_t)bc*HWX + t] = acc;
}

// Branch gating: global-avg-pool -> FC(96) gelu -> FC(3) -> softmax. 1 block per batch.
__global__ void k_gate(const float* __restrict__ e1, const float* __restrict__ w1,
                       const float* __restrict__ b1, const float* __restrict__ w2,
                       const float* __restrict__ b2, float* __restrict__ gw)
{
  int bi = blockIdx.x, t = threadIdx.x;      // 384 threads
  __shared__ float pooled[DIMC]; __shared__ float hid[GATEHID]; __shared__ float lg[3];
  { const float* src = e1 + (size_t)bi*DIMC*HWX + (size_t)t*HWX;
    float s = 0.f; for (int i = 0; i < HWX; ++i) s += src[i];
    pooled[t] = s * (1.0f/HWX); }
  __syncthreads();
  if (t < GATEHID) { float s = b1[t]; const float* wr = w1 + (size_t)t*DIMC;
    for (int i = 0; i < DIMC; ++i) s += wr[i]*pooled[i]; hid[t] = gelu_f(s); }
  __syncthreads();
  if (t < 3) { float s = b2[t]; const float* wr = w2 + (size_t)t*GATEHID;
    for (int i = 0; i < GATEHID; ++i) s += wr[i]*hid[i]; lg[t] = s; }
  __syncthreads();
  if (t == 0) {
    float mx = fmaxf(lg[0], fmaxf(lg[1], lg[2]));
    float e0 = expf(lg[0]-mx), ee1 = expf(lg[1]-mx), e2 = expf(lg[2]-mx);
    float s = e0+ee1+e2;
    gw[bi*3+0] = e0/s; gw[bi*3+1] = ee1/s; gw[bi*3+2] = e2/s;
  }
}

__global__ void k_wsum(const float* b0, const float* b1, const float* b2,
                       const float* __restrict__ gw, float* out)
{
  size_t i = (size_t)blockIdx.x*blockDim.x + threadIdx.x;
  if (i >= (size_t)BATCH*DIMC*HWX) return;
  int bi = (int)(i / (DIMC*HWX));
  out[i] = gw[bi*3]*b0[i] + gw[bi*3+1]*b1[i] + gw[bi*3+2]*b2[i];
}

// ------------------------------ token path ---------------------------------
__global__ void k_tok(const float* __restrict__ x, float* __restrict__ tok)
{
  size_t i = (size_t)blockIdx.x*blockDim.x + threadIdx.x;
  if (i >= (size_t)TOK*DIMC) return;
  int row = (int)(i / DIMC), c = (int)(i % DIMC);
  int bi = row / LSEQ, l = row % LSEQ;
  tok[i] = x[(size_t)bi*DIMC*HWX + (size_t)c*HWX + l];
}

__global__ void k_ln(const float* __restrict__ in, const float* __restrict__ w,
                     const float* __restrict__ b, float* __restrict__ out)
{
  int row = blockIdx.x;
  const float* x = in + (size_t)row*DIMC;
  __shared__ float s1[128], s2[128];
  float a = 0.f, a2 = 0.f;
  for (int i = threadIdx.x; i < DIMC; i += 128) { float v = x[i]; a += v; a2 += v*v; }
  s1[threadIdx.x] = a; s2[threadIdx.x] = a2; __syncthreads();
  for (int st = 64; st > 0; st >>= 1) {
    if ((int)threadIdx.x < st) { s1[threadIdx.x] += s1[threadIdx.x+st]; s2[threadIdx.x] += s2[threadIdx.x+st]; }
    __syncthreads();
  }
  float m = s1[0]/DIMC, inv = rsqrtf(s2[0]/DIMC - m*m + 1e-5f);
  float* o = out + (size_t)row*DIMC;
  for (int i = threadIdx.x; i < DIMC; i += 128) o[i] = (x[i]-m)*inv*w[i] + b[i];
}

__global__ void k_flip(const float* __restrict__ in, float* __restrict__ out)
{
  size_t i = (size_t)blockIdx.x*blockDim.x + threadIdx.x;
  if (i >= (size_t)TOK*DIMC) return;
  int row = (int)(i / DIMC), c = (int)(i % DIMC);
  int bi = row / LSEQ, l = row % LSEQ;
  out[i] = in[((size_t)bi*LSEQ + (LSEQ-1-l))*DIMC + c];
}

// ------------------------------- mamba -------------------------------------
__global__ void k_dtda(const float* __restrict__ zx, const float* __restrict__ dtb,
                       const float* __restrict__ alog, float* __restrict__ dt,
                       float* __restrict__ dA)
{
  int i = blockIdx.x*blockDim.x + threadIdx.x;
  if (i >= TOK*NHEADS) return;
  int row = i / NHEADS, h = i % NHEADS;
  float x = zx[(size_t)row*NPROJ + 2*DINNER + 2*DSTATE + h] + dtb[h];
  float sp = (x > 20.f) ? x : log1pf(expf(x));
  dt[i] = sp;
  dA[i] = expf(-expf(alog[h]) * sp);
}

__global__ void k_convsilu(const float* __restrict__ zx, const float* __restrict__ cw,
                           const float* __restrict__ cb, float* __restrict__ out)
{
  size_t i = (size_t)blockIdx.x*blockDim.x + threadIdx.x;
  if (i >= (size_t)TOK*XBCW) return;
  int row = (int)(i / XBCW), c = (int)(i % XBCW);
  int bi = row / LSEQ, l = row % LSEQ;
  const float* wr = cw + (size_t)c*4;
  float acc = cb[c];
  #pragma unroll
  for (int j = 0; j < 4; ++j) {
    int ls = l - 3 + j;
    if (ls >= 0) acc += wr[j] * zx[((size_t)bi*LSEQ + ls)*NPROJ + DINNER + c];
  }
  out[i] = silu_f(acc);
}

// Selective scan: 1 block per (b, head), 64 lanes = HEADDIM; each lane keeps its
// 64-entry state row in VGPRs. B_t / C_t staged via LDS; next row prefetched
// (global_prefetch_b8) to hide L2/HBM latency across the serial t-loop.
__global__ void k_scan(const float* __restrict__ xbc, const float* __restrict__ dt,
                       const float* __restrict__ dA, const float* __restrict__ Dp,
                       float* __restrict__ y)
{
  int bh = blockIdx.x, bi = bh / NHEADS, hd = bh % NHEADS;
  int p = threadIdx.x;                       // 0..63
  __shared__ float Bs[DSTATE], Cs[DSTATE];
  float h[DSTATE];
  #pragma unroll
  for (int n = 0; n < DSTATE; ++n) h[n] = 0.f;
  float Dv = Dp[hd];
  for (int t = 0; t < LSEQ; ++t) {
    size_t row = (size_t)bi*LSEQ + t;
    const float* xr = xbc + row*XBCW;
    if (t + 1 < LSEQ) {                      // prefetch next timestep's row
      __builtin_prefetch(xr + XBCW + DINNER + p, 0, 0);
      __builtin_prefetch(xr + XBCW + hd*HEADDIM + p, 0, 0);
    }
    Bs[p] = xr[DINNER + p];
    Cs[p] = xr[DINNER + DSTATE + p];
    float xt  = xr[hd*HEADDIM + p];
    float dtv = dt[row*NHEADS + hd];
    float dAv = dA[row*NHEADS + hd];
    __syncthreads();
    float dtx = dtv * xt, acc = 0.f;
    #pragma unroll
    for (int n = 0; n < DSTATE; ++n) { h[n] = dAv*h[n] + Bs[n]*dtx; acc += h[n]*Cs[n]; }
    y[row*DINNER + hd*HEADDIM + p] = acc + Dv*xt;
    __syncthreads();
  }
}

__global__ void k_zmul(float* __restrict__ y, const float* __restrict__ zx)
{
  size_t i = (size_t)blockIdx.x*blockDim.x + threadIdx.x;
  if (i >= (size_t)TOK*DINNER) return;
  int row = (int)(i / DINNER), c = (int)(i % DINNER);
  y[i] *= silu_f(zx[(size_t)row*NPROJ + c]);
}

// -------------------------- merge / residual / MLP -------------------------
__global__ void k_concat(const float* __restrict__ fwd, const float* __restrict__ bwd,
                         float* __restrict__ cat)
{
  size_t i = (size_t)blockIdx.x*blockDim.x + threadIdx.x;
  if (i >= (size_t)TOK*2*DIMC) return;
  int row = (int)(i / (2*DIMC)), c = (int)(i % (2*DIMC));
  if (c < DIMC) cat[i] = fwd[(size_t)row*DIMC + c];
  else {
    int bi = row / LSEQ, l = row % LSEQ;
    cat[i] = bwd[((size_t)bi*LSEQ + (LSEQ-1-l))*DIMC + (c-DIMC)];
  }
}

__global__ void k_add3(const float* __restrict__ tok, const float* __restrict__ loc,
                       const float* __restrict__ mix, float* __restrict__ out)
{
  size_t i = (size_t)blockIdx.x*blockDim.x + threadIdx.x;
  if (i >= (size_t)TOK*DIMC) return;
  int row = (int)(i / DIMC), c = (int)(i % DIMC);
  int bi = row / LSEQ, l = row % LSEQ;
  out[i] = tok[i] + mix[i] + loc[(size_t)bi*DIMC*HWX + (size_t)c*HWX + l];
}

__global__ void k_gelu_ew(float* __restrict__ x, size_t n)
{
  size_t i = (size_t)blockIdx.x*blockDim.x + threadIdx.x;
  if (i < n) x[i] = gelu_f(x[i]);
}

__global__ void k_final(const float* __restrict__ t, const float* __restrict__ m,
                        float* __restrict__ out)
{
  size_t i = (size_t)blockIdx.x*blockDim.x + threadIdx.x;
  if (i >= (size_t)TOK*DIMC) return;
  int row = (int)(i / DIMC), c = (int)(i % DIMC);
  int bi = row / LSEQ, l = row % LSEQ;
  out[(size_t)bi*DIMC*HWX + (size_t)c*HWX + l] = t[i] + m[i];
}

// ---------------------------------------------------------------------------
static inline void gemm(int bmode, const float* A, const float* Bm, const float* bias,
                        float* C, int M, int N, int K, int batch,
                        long sA, long sB, long sC, hipStream_t s)
{
  long tiles = (long)((M+15)/16) * ((N+63)/64);
  int blocks = (int)((tiles + 3) / 4);
  dim3 g(blocks, 1, batch);
  if (bmode == 0) k_gemm<0><<<g, 128, 0, s>>>(A, Bm, bias, C, M, N, K, sA, sB, sC);
  else            k_gemm<1><<<g, 128, 0, s>>>(A, Bm, bias, C, M, N, K, sA, sB, sC);
}

#define EWGRID(n) dim3((unsigned)(((n)+255)/256)), dim3(256)

extern "C" void kernel_launch(void* const* d_in, const int* in_sizes, int n_in,
                              void* d_out, int out_size, void* d_ws, size_t ws_size,
                              hipStream_t stream)
{
  // ---- inputs, flattened in setup_inputs() dict order ----
  const float* x        = (const float*)d_in[0];
  const float* exp_w    = (const float*)d_in[1];
  const float* exp_gn_w = (const float*)d_in[2];
  const float* exp_gn_b = (const float*)d_in[3];
  const float* br_w[3]  = {(const float*)d_in[4], (const float*)d_in[5], (const float*)d_in[6]};
  const float* br_gn_w[3]={(const float*)d_in[7], (const float*)d_in[8], (const float*)d_in[9]};
  const float* br_gn_b[3]={(const float*)d_in[10],(const float*)d_in[11],(const float*)d_in[12]};
  const float* gate_w1  = (const float*)d_in[13];
  const float* gate_b1  = (const float*)d_in[14];
  const float* gate_w2  = (const float*)d_in[15];
  const float* gate_b2  = (const float*)d_in[16];
  const float* proj_w   = (const float*)d_in[17];
  const float* proj_gn_w= (const float*)d_in[18];
  const float* proj_gn_b= (const float*)d_in[19];
  const float* ln_w     = (const float*)d_in[20];
  const float* ln_b     = (const float*)d_in[21];
  // m_fwd: 22..28, m_bwd: 29..35
  const float* merge_w  = (const float*)d_in[36];
  const float* merge_b  = (const float*)d_in[37];
  const float* mlp_ln_w = (const float*)d_in[38];
  const float* mlp_ln_b = (const float*)d_in[39];
  const float* fc1_w    = (const float*)d_in[40];
  const float* fc1_b    = (const float*)d_in[41];
  const float* fc2_w    = (const float*)d_in[42];
  const float* fc2_b    = (const float*)d_in[43];

  // ---- workspace bump allocator ----
  size_t off = 0;
  auto alloc = [&](size_t nf) -> float* {
    float* p = (float*)((char*)d_ws + off);
    off = (off + nf*sizeof(float) + 255) & ~(size_t)255;
    return p;
  };
  const size_t BCHW = (size_t)BATCH*DIMC*HWX;   // 884736
  const size_t TC   = (size_t)TOK*DIMC;         // 884736
  float* e1     = alloc(BCHW);
  float* br0    = alloc(BCHW);
  float* br1    = alloc(BCHW);
  float* br2    = alloc(BCHW);
  float* gw     = alloc(16);
  float* x2     = alloc(BCHW);
  float* localb = alloc(BCHW);
  float* tokens = alloc(TC);
  float* normed = alloc(TC);
  float* ubwd   = alloc(TC);
  float* fwdout = alloc(TC);
  float* bwdout = alloc(TC);
  float* catb   = alloc((size_t)TOK*2*DIMC);
  float* mixed  = alloc(TC);
  float* tbuf   = alloc(TC);
  float* mln    = alloc(TC);
  float* hid    = alloc((size_t)TOK*MLPHID);
  float* zx     = alloc((size_t)TOK*NPROJ);
  float* dtb_s  = alloc((size_t)TOK*NHEADS);
  float* dab_s  = alloc((size_t)TOK*NHEADS);
  float* xbc    = alloc((size_t)TOK*XBCW);
  float* yb     = alloc((size_t)TOK*DINNER);
  (void)ws_size; (void)in_sizes; (void)n_in; (void)out_size;

  // ================= local enhancer =================
  gemm(0, exp_w, x, nullptr, e1, DIMC, HWX, DIMC, BATCH, 0, (long)DIMC*HWX, (long)DIMC*HWX, stream);
  k_gn_gelu<<<BATCH*GROUPS, 256, 0, stream>>>(e1, exp_gn_w, exp_gn_b);
  const int ks[3] = {1, 3, 5};
  float* brp[3] = {br0, br1, br2};
  for (int i = 0; i < 3; ++i) {
    k_dwconv<<<BATCH*DIMC, HWX, 0, stream>>>(e1, br_w[i], brp[i], ks[i]);
    k_gn_gelu<<<BATCH*GROUPS, 256, 0, stream>>>(brp[i], br_gn_w[i], br_gn_b[i]);
  }
  k_gate<<<BATCH, DIMC, 0, stream>>>(e1, gate_w1, gate_b1, gate_w2, gate_b2, gw);
  k_wsum<<<EWGRID(BCHW), 0, stream>>>(br0, br1, br2, gw, x2);
  gemm(0, proj_w, x2, nullptr, localb, DIMC, HWX, DIMC, BATCH, 0, (long)DIMC*HWX, (long)DIMC*HWX, stream);
  k_gn_gelu<<<BATCH*GROUPS, 256, 0, stream>>>(localb, proj_gn_w, proj_gn_b);

  // ================= token path =================
  k_tok<<<EWGRID(TC), 0, stream>>>(x, tokens);
  k_ln<<<TOK, 128, 0, stream>>>(tokens, ln_w, ln_b, normed);
  k_flip<<<EWGRID(TC), 0, stream>>>(normed, ubwd);

  // ================= mamba (fwd then bwd, reusing big scratch) =================
  for (int dir = 0; dir < 2; ++dir) {
    int base = 22 + dir*7;
    const float* in_w    = (const float*)d_in[base+0];
    const float* conv_w  = (const float*)d_in[base+1];
    const float* conv_b  = (const float*)d_in[base+2];
    const float* dt_bias = (const float*)d_in[base+3];
    const float* A_log   = (const float*)d_in[base+4];
    const float* Dm      = (const float*)d_in[base+5];
    const float* out_w   = (const float*)d_in[base+6];
    const float* u   = dir ? ubwd : normed;
    float*       dst = dir ? bwdout : fwdout;

    gemm(1, u, in_w, nullptr, zx, TOK, NPROJ, DIMC, 1, 0, 0, 0, stream);
    k_dtda<<<(TOK*NHEADS + 255)/256, 256, 0, stream>>>(zx, dt_bias, A_log, dtb_s, dab_s);
    k_convsilu<<<EWGRID((size_t)TOK*XBCW), 0, stream>>>(zx, conv_w, conv_b, xbc);
    k_scan<<<BATCH*NHEADS, HEADDIM, 0, stream>>>(xbc, dtb_s, dab_s, Dm, yb);
    k_zmul<<<EWGRID((size_t)TOK*DINNER), 0, stream>>>(yb, zx);
    gemm(1, yb, out_w, nullptr, dst, TOK, DIMC, DINNER, 1, 0, 0, 0, stream);
  }

  // ================= merge + residual + MLP =================
  k_concat<<<EWGRID((size_t)TOK*2*DIMC), 0, stream>>>(fwdout, bwdout, catb);
  gemm(1, catb, merge_w, merge_b, mixed, TOK, DIMC, 2*DIMC, 1, 0, 0, 0, stream);
  k_add3<<<EWGRID(TC), 0, stream>>>(tokens, localb, mixed, tbuf);
  k_ln<<<TOK, 128, 0, stream>>>(tbuf, mlp_ln_w, mlp_ln_b, mln);
  gemm(1, mln, fc1_w, fc1_b, hid, TOK, MLPHID, DIMC, 1, 0, 0, 0, stream);
  k_gelu_ew<<<EWGRID((size_t)TOK*MLPHID), 0, stream>>>(hid, (size_t)TOK*MLPHID);
  gemm(1, hid, fc2_w, fc2_b, mln, TOK, DIMC, MLPHID, 1, 0, 0, 0, stream);  // mln reused as mlp out
  k_final<<<EWGRID(TC), 0, stream>>>(tbuf, mln, (float*)d_out);
}